// TopLineEncoder_16415365005453
// MI455X (gfx1250) — compile-verified
//
#include <hip/hip_runtime.h>
#include <hip/hip_bf16.h>
#include <math.h>

#define NPOS  80
#define NCHAR 256
#define HDIM  512
#define IDIM  (NPOS * NCHAR)     // 20480
#define BATCH 8192
#define LN_EPS 1e-5f
#define KC    32                 // GEMM K-chunk staged in LDS
#define LDP   34                 // padded LDS row stride (floats): conflict-free, 8B-aligned

typedef __attribute__((ext_vector_type(2))) float v2f;
typedef __attribute__((ext_vector_type(8))) float v8f;

// ---------------------------------------------------------------------------
// Kernel 1: transpose W1 (HDIM x IDIM, row-major) -> W1T (IDIM x HDIM)
// so the embedding gather becomes contiguous 2KB rows.
// ---------------------------------------------------------------------------
__global__ void k_transpose_w1(const float* __restrict__ W1,
                               float* __restrict__ W1T) {
  __shared__ float tile[32][33];              // +1 pad: no LDS bank conflicts
  const int i0 = blockIdx.x * 32;             // IDIM index
  const int j0 = blockIdx.y * 32;             // HDIM index
  const int tx = threadIdx.x;                 // 0..31
  const int ty = threadIdx.y;                 // 0..7
#pragma unroll
  for (int r = ty; r < 32; r += 8)
    tile[r][tx] = W1[(size_t)(j0 + r) * IDIM + (i0 + tx)];
  __syncthreads();
#pragma unroll
  for (int r = ty; r < 32; r += 8)
    W1T[(size_t)(i0 + r) * HDIM + (j0 + tx)] = tile[tx][r];
}

// ---------------------------------------------------------------------------
// Block-wide LayerNorm stats over 512 values (256 threads x 2 values).
// Returns {mu, rsigma}. wave32 shuffle reduce + 8-wave LDS combine.
// ---------------------------------------------------------------------------
__device__ __forceinline__ float2 ln_stats_512(float a0, float a1) {
  float s = a0 + a1;
  float q = a0 * a0 + a1 * a1;
#pragma unroll
  for (int off = 16; off > 0; off >>= 1) {
    s += __shfl_xor(s, off, 32);
    q += __shfl_xor(q, off, 32);
  }
  __shared__ float ssum[8], sq[8], stats[2];
  const int wid  = threadIdx.x >> 5;
  const int lane = threadIdx.x & 31;
  if (lane == 0) { ssum[wid] = s; sq[wid] = q; }
  __syncthreads();
  if (threadIdx.x == 0) {
    float S = 0.f, Q = 0.f;
#pragma unroll
    for (int w = 0; w < 8; ++w) { S += ssum[w]; Q += sq[w]; }
    const float mu  = S * (1.0f / (float)HDIM);
    const float var = Q * (1.0f / (float)HDIM) - mu * mu;
    stats[0] = mu;
    stats[1] = rsqrtf(var + LN_EPS);
  }
  __syncthreads();
  return make_float2(stats[0], stats[1]);
}

__device__ __forceinline__ float elu(float x) {
  return x > 0.f ? x : expm1f(x);
}

// ---------------------------------------------------------------------------
// Kernel 2: embedding gather-sum (+b1) -> LayerNorm -> ELU -> h1[B][H]
// One 256-thread block per batch row. Each thread owns columns {2*tid, 2*tid+1}
// -> one 8B load per gathered row; block reads a contiguous 2KB row of W1T.
// Next column is prefetched (global_prefetch_b8) to hide L2 latency on the
// 80-deep serial gather chain.
// ---------------------------------------------------------------------------
__global__ void k_embed_ln1(const int* __restrict__ msg,
                            const float* __restrict__ W1T,
                            const float* __restrict__ b1,
                            const float* __restrict__ g1,
                            const float* __restrict__ be1,
                            float* __restrict__ h1) {
  const int b   = blockIdx.x;
  const int j0  = threadIdx.x * 2;
  const int* m  = msg + (size_t)b * NPOS;
  float a0 = 0.f, a1 = 0.f;
  int c = m[0];
  for (int p = 0; p < NPOS; ++p) {
    const float* col = W1T + (size_t)(p * NCHAR + c) * HDIM;
    if (p + 1 < NPOS) {
      c = m[p + 1];                              // uniform per block
      __builtin_prefetch(W1T + (size_t)((p + 1) * NCHAR + c) * HDIM + j0, 0, 1);
    }
    const v2f v = *(const v2f*)(col + j0);
    a0 += v.x;
    a1 += v.y;
  }
  a0 += b1[j0];
  a1 += b1[j0 + 1];
  const float2 st = ln_stats_512(a0, a1);
  const float v0 = (a0 - st.x) * st.y * g1[j0]     + be1[j0];
  const float v1 = (a1 - st.x) * st.y * g1[j0 + 1] + be1[j0 + 1];
  h1[(size_t)b * HDIM + j0]     = elu(v0);
  h1[(size_t)b * HDIM + j0 + 1] = elu(v1);
}

// Fallback when workspace can't hold W1T: gather straight from W1 (strided).
__global__ void k_embed_ln1_direct(const int* __restrict__ msg,
                                   const float* __restrict__ W1,
                                   const float* __restrict__ b1,
                                   const float* __restrict__ g1,
                                   const float* __restrict__ be1,
                                   float* __restrict__ h1) {
  const int b   = blockIdx.x;
  const int tid = threadIdx.x;
  const int* m  = msg + (size_t)b * NPOS;
  float a0 = 0.f, a1 = 0.f;
  for (int p = 0; p < NPOS; ++p) {
    const size_t idx = (size_t)p * NCHAR + m[p];
    a0 += W1[(size_t)tid * IDIM + idx];
    a1 += W1[(size_t)(tid + 256) * IDIM + idx];
  }
  a0 += b1[tid];
  a1 += b1[tid + 256];
  const float2 st = ln_stats_512(a0, a1);
  const float v0 = (a0 - st.x) * st.y * g1[tid]       + be1[tid];
  const float v1 = (a1 - st.x) * st.y * g1[tid + 256] + be1[tid + 256];
  h1[(size_t)b * HDIM + tid]       = elu(v0);
  h1[(size_t)b * HDIM + tid + 256] = elu(v1);
}

// ---------------------------------------------------------------------------
// Kernel 3: h2 = h1 @ W2^T + b2 via V_WMMA_F32_16X16X4_F32 (exact fp32),
// LDS-staged. Block = 256 threads (8 waves), tile M=128 x N=64.
// Per K-chunk (KC=32): coalesced float4 global loads stage
//   sA: 128x32 (stride LDP=34)   sB: 64x32 (stride 34)
// then each wave reads conflict-free ds_load_b64 fragments:
//   A(16x4):  lane%16 = M row, (lane/16)*2 + r = K          (2 VGPRs)
//   B(4x16):  lane%16 = N col, (lane/16)*2 + r = K          (2 VGPRs)
//   D(16x16): lane%16 = N col, VGPR r -> M = r + (lane/16)*8 (8 VGPRs)
// B is loaded from global once per block instead of once per wave (8x reuse).
// ---------------------------------------------------------------------------
__global__ void k_gemm_wmma(const float* __restrict__ h1,
                            const float* __restrict__ W2,
                            const float* __restrict__ b2,
                            float* __restrict__ h2) {
  __shared__ float sA[128 * LDP];   // 17408 B
  __shared__ float sB[64 * LDP];    //  8704 B
  const int tid   = threadIdx.x;    // 0..255
  const int wid   = tid >> 5;       // 0..7
  const int lane  = tid & 31;
  const int m0    = blockIdx.x * 128;
  const int n0    = blockIdx.y * 64;
  const int mrow  = lane & 15;
  const int khalf = (lane >> 4) << 1;          // 0 or 2

  v8f acc0 = {}, acc1 = {}, acc2 = {}, acc3 = {};

  for (int k0 = 0; k0 < HDIM; k0 += KC) {
    // ---- stage A: 128 x 32 floats = 1024 float4, 4 per thread ----
#pragma unroll
    for (int i = 0; i < 4; ++i) {
      const int idx = tid + i * 256;           // float4 index
      const int r   = idx >> 3;                // 8 float4 per row
      const int cc  = (idx & 7) << 2;          // 0,4,..,28
      const float4 v = *(const float4*)(h1 + (size_t)(m0 + r) * HDIM + k0 + cc);
      v2f lo = {v.x, v.y};
      v2f hi = {v.z, v.w};
      *(v2f*)(sA + r * LDP + cc)     = lo;
      *(v2f*)(sA + r * LDP + cc + 2) = hi;
    }
    // ---- stage B: 64 x 32 floats = 512 float4, 2 per thread ----
#pragma unroll
    for (int i = 0; i < 2; ++i) {
      const int idx = tid + i * 256;
      const int r   = idx >> 3;
      const int cc  = (idx & 7) << 2;
      const float4 v = *(const float4*)(W2 + (size_t)(n0 + r) * HDIM + k0 + cc);
      v2f lo = {v.x, v.y};
      v2f hi = {v.z, v.w};
      *(v2f*)(sB + r * LDP + cc)     = lo;
      *(v2f*)(sB + r * LDP + cc + 2) = hi;
    }
    __syncthreads();

    const float* Aw = sA + (wid * 16 + mrow) * LDP + khalf;
    const float* Bw = sB + mrow * LDP + khalf;
#pragma unroll
    for (int kk = 0; kk < KC; kk += 4) {
      const v2f a   = *(const v2f*)(Aw + kk);
      const v2f bb0 = *(const v2f*)(Bw + kk);
      const v2f bb1 = *(const v2f*)(Bw + 16 * LDP + kk);
      const v2f bb2 = *(const v2f*)(Bw + 32 * LDP + kk);
      const v2f bb3 = *(const v2f*)(Bw + 48 * LDP + kk);
      acc0 = __builtin_amdgcn_wmma_f32_16x16x4_f32(false, a, false, bb0, (short)0, acc0, false, false);
      acc1 = __builtin_amdgcn_wmma_f32_16x16x4_f32(false, a, false, bb1, (short)0, acc1, false, false);
      acc2 = __builtin_amdgcn_wmma_f32_16x16x4_f32(false, a, false, bb2, (short)0, acc2, false, false);
      acc3 = __builtin_amdgcn_wmma_f32_16x16x4_f32(false, a, false, bb3, (short)0, acc3, false, false);
    }
    __syncthreads();
  }

  const int ncol  = lane & 15;
  const int mbase = (lane >> 4) << 3;          // 0 or 8
  float* out = h2 + (size_t)(m0 + wid * 16 + mbase) * HDIM + n0;
  const float bj0 = b2[n0 + ncol];
  const float bj1 = b2[n0 + 16 + ncol];
  const float bj2 = b2[n0 + 32 + ncol];
  const float bj3 = b2[n0 + 48 + ncol];
#pragma unroll
  for (int r = 0; r < 8; ++r) {
    float* row = out + (size_t)r * HDIM;
    row[ncol]      = acc0[r] + bj0;
    row[16 + ncol] = acc1[r] + bj1;
    row[32 + ncol] = acc2[r] + bj2;
    row[48 + ncol] = acc3[r] + bj3;
  }
}

// ---------------------------------------------------------------------------
// Kernel 4: final LayerNorm + ELU -> d_out
// ---------------------------------------------------------------------------
__global__ void k_ln2_elu(const float* __restrict__ h2,
                          const float* __restrict__ g2,
                          const float* __restrict__ be2,
                          float* __restrict__ out) {
  const int b  = blockIdx.x;
  const int j0 = threadIdx.x * 2;
  const v2f v  = *(const v2f*)(h2 + (size_t)b * HDIM + j0);
  const float a0 = v.x, a1 = v.y;
  const float2 st = ln_stats_512(a0, a1);
  const float v0 = (a0 - st.x) * st.y * g2[j0]     + be2[j0];
  const float v1 = (a1 - st.x) * st.y * g2[j0 + 1] + be2[j0 + 1];
  v2f o = {elu(v0), elu(v1)};
  *(v2f*)(out + (size_t)b * HDIM + j0) = o;
}

// ---------------------------------------------------------------------------
extern "C" void kernel_launch(void* const* d_in, const int* in_sizes, int n_in,
                              void* d_out, int out_size, void* d_ws, size_t ws_size,
                              hipStream_t stream) {
  (void)in_sizes; (void)n_in; (void)out_size;
  const int*   msg = (const int*)  d_in[0];
  const float* W1  = (const float*)d_in[1];
  const float* b1  = (const float*)d_in[2];
  const float* g1  = (const float*)d_in[3];
  const float* be1 = (const float*)d_in[4];
  const float* W2  = (const float*)d_in[5];
  const float* b2  = (const float*)d_in[6];
  const float* g2  = (const float*)d_in[7];
  const float* be2 = (const float*)d_in[8];
  float* out = (float*)d_out;

  const size_t w1t_bytes = (size_t)IDIM * HDIM * sizeof(float);   // 42 MB
  const size_t h_bytes   = (size_t)BATCH * HDIM * sizeof(float);  // 16.8 MB
  char* ws = (char*)d_ws;
  const bool use_transpose = ws_size >= (w1t_bytes + 2 * h_bytes);

  float* h1;
  float* h2;
  if (use_transpose) {
    float* W1T = (float*)ws;
    h1 = (float*)(ws + w1t_bytes);
    h2 = (float*)(ws + w1t_bytes + h_bytes);
    k_transpose_w1<<<dim3(IDIM / 32, HDIM / 32), dim3(32, 8), 0, stream>>>(W1, W1T);
    k_embed_ln1<<<BATCH, 256, 0, stream>>>(msg, W1T, b1, g1, be1, h1);
  } else {
    h1 = (float*)ws;
    h2 = (float*)(ws + h_bytes);
    k_embed_ln1_direct<<<BATCH, 256, 0, stream>>>(msg, W1, b1, g1, be1, h1);
  }

  k_gemm_wmma<<<dim3(BATCH / 128, HDIM / 64), 256, 0, stream>>>(h1, W2, b2, h2);
  k_ln2_elu<<<BATCH, 256, 0, stream>>>(h2, g2, be2, out);
}